// GATLayer_12532714569906
// MI455X (gfx1250) — compile-verified
//
#include <hip/hip_runtime.h>
#include <hip/hip_bf16.h>

#define HD 128            // H*D = 4*32
#define IN_DIM 64
#define NHEAD 4
#define DHEAD 32
#define NEG_SLOPE 0.1f

typedef __attribute__((ext_vector_type(2))) float v2f;
typedef __attribute__((ext_vector_type(8))) float v8f;

// ---- monotonic float <-> uint encoding for atomic max on floats ----
__device__ __forceinline__ unsigned fenc(float f) {
  unsigned u = __float_as_uint(f);
  return (f >= 0.0f) ? (u | 0x80000000u) : ~u;
}
__device__ __forceinline__ float fdec(unsigned u) {
  return (u & 0x80000000u) ? __uint_as_float(u & 0x7FFFFFFFu) : __uint_as_float(~u);
}

__device__ __forceinline__ float lrelu(float e) {
  return e >= 0.0f ? e : NEG_SLOPE * e;
}

// =====================================================================
// Kernel 1: feat = x @ W_fc  via V_WMMA_F32_16X16X4_F32.
// One wave32 computes a 16-row x 128-col tile (16 K-steps x 8 col tiles
// = 128 WMMA ops). fp32 WMMA keeps reference precision.
// =====================================================================
__global__ void __launch_bounds__(256)
gat_proj_wmma(const float* __restrict__ x, const float* __restrict__ W,
              float* __restrict__ feat, int nrows) {
  const int wave = (blockIdx.x * blockDim.x + threadIdx.x) >> 5;
  const int lane = threadIdx.x & 31;
  const int rowbase = wave * 16;
  if (rowbase >= nrows) return;               // uniform per wave

  const int mrow  = lane & 15;                // M index for A, N index for B/C
  const int khalf = (lane >> 4) * 2;          // lanes 16-31 hold K+2/K+3

#if __has_builtin(__builtin_amdgcn_wmma_f32_16x16x4_f32)
  int arow = rowbase + mrow;
  if (arow >= nrows) arow = nrows - 1;        // data clamp, no divergence
  const float* __restrict__ xrow = x + arow * IN_DIM;

  v8f acc[8] = {};                            // 8 col tiles of 16

  #pragma unroll
  for (int k0 = 0; k0 < IN_DIM; k0 += 4) {
    v2f a;
    a.x = xrow[k0 + khalf + 0];
    a.y = xrow[k0 + khalf + 1];
    #pragma unroll
    for (int t = 0; t < 8; ++t) {
      const int col = t * 16 + mrow;
      v2f b;
      b.x = W[(k0 + khalf + 0) * HD + col];
      b.y = W[(k0 + khalf + 1) * HD + col];
      acc[t] = __builtin_amdgcn_wmma_f32_16x16x4_f32(
          /*neg_a=*/false, a, /*neg_b=*/false, b,
          /*c_mod=*/(short)0, acc[t], /*reuse_a=*/false, /*reuse_b=*/false);
    }
  }

  // C/D layout: VGPR i holds (M = i + 8*(lane>>4), N = lane&15)
  const int moff = (lane >> 4) * 8;
  const int ncol = lane & 15;
  #pragma unroll
  for (int t = 0; t < 8; ++t) {
    #pragma unroll
    for (int i = 0; i < 8; ++i) {
      const int node = rowbase + moff + i;
      if (node < nrows) feat[node * HD + t * 16 + ncol] = acc[t][i];
    }
  }
#else
  // scalar fallback (should not be taken on gfx1250)
  for (int idx = lane; idx < 16 * HD; idx += 32) {
    const int r = rowbase + (idx >> 7);
    const int c = idx & (HD - 1);
    if (r < nrows) {
      float s = 0.0f;
      for (int k = 0; k < IN_DIM; ++k) s += x[r * IN_DIM + k] * W[k * HD + c];
      feat[r * HD + c] = s;
    }
  }
#endif
}

// =====================================================================
// Kernel 2: el/er per (node, head): dot(feat[n,h,:], attn_{l,r}[h,:])
// =====================================================================
__global__ void __launch_bounds__(256)
gat_attn_dots(const float* __restrict__ feat,
              const float* __restrict__ attn_l, const float* __restrict__ attn_r,
              float* __restrict__ el, float* __restrict__ er, int n_nh) {
  const int tid = blockIdx.x * blockDim.x + threadIdx.x;
  if (tid >= n_nh) return;
  const int h = tid & (NHEAD - 1);
  const int node = tid >> 2;
  const float* __restrict__ f = feat + node * HD + h * DHEAD;
  const float* __restrict__ al = attn_l + h * DHEAD;
  const float* __restrict__ ar = attn_r + h * DHEAD;
  float sl = 0.0f, sr = 0.0f;
  #pragma unroll
  for (int d = 0; d < DHEAD; ++d) {
    const float v = f[d];
    sl += v * al[d];
    sr += v * ar[d];
  }
  el[tid] = sl;
  er[tid] = sr;
}

// =====================================================================
// Kernel 3: pass 1 — segment max over dst via u32 atomic max (encoded)
// one thread per (edge, head)
// =====================================================================
__global__ void __launch_bounds__(256)
gat_edge_max(const int* __restrict__ src, const int* __restrict__ dst,
             const float* __restrict__ w, const float* __restrict__ el,
             const float* __restrict__ er, unsigned* __restrict__ menc, int n_eh) {
  const int tid = blockIdx.x * blockDim.x + threadIdx.x;
  if (tid >= n_eh) return;
  const int h = tid & (NHEAD - 1);
  const int e = tid >> 2;
  const int s = src[e], d = dst[e];
  float v = lrelu(el[s * NHEAD + h] + er[d * NHEAD + h]) * w[e];
  atomicMax(&menc[d * NHEAD + h], fenc(v));
}

// =====================================================================
// Kernel 4: pass 2 — den[dst,h] += exp(e - m[dst,h])
// =====================================================================
__global__ void __launch_bounds__(256)
gat_edge_den(const int* __restrict__ src, const int* __restrict__ dst,
             const float* __restrict__ w, const float* __restrict__ el,
             const float* __restrict__ er, const unsigned* __restrict__ menc,
             float* __restrict__ den, int n_eh) {
  const int tid = blockIdx.x * blockDim.x + threadIdx.x;
  if (tid >= n_eh) return;
  const int h = tid & (NHEAD - 1);
  const int e = tid >> 2;
  const int s = src[e], d = dst[e];
  float v = lrelu(el[s * NHEAD + h] + er[d * NHEAD + h]) * w[e];
  const float m = fdec(menc[d * NHEAD + h]);
  unsafeAtomicAdd(&den[d * NHEAD + h], __expf(v - m));
}

// =====================================================================
// Kernel 5: pass 3 — out[dst] += a * feat[src]; one wave32 per edge.
// Lane l handles components [4l, 4l+4) (float4); head = l>>3; the
// softmax coefficient is computed in lanes 0..3 and shfl-broadcast.
// =====================================================================
__global__ void __launch_bounds__(256)
gat_edge_scatter(const int* __restrict__ src, const int* __restrict__ dst,
                 const float* __restrict__ w, const float* __restrict__ el,
                 const float* __restrict__ er, const unsigned* __restrict__ menc,
                 const float* __restrict__ den, const float* __restrict__ feat,
                 float* __restrict__ out, int n_edges) {
  const int warp = (blockIdx.x * blockDim.x + threadIdx.x) >> 5;
  const int lane = threadIdx.x & 31;
  if (warp >= n_edges) return;                // uniform per wave
  const int s = src[warp], d = dst[warp];
  const float wv = w[warp];

  // every lane computes 'a' for head = lane&3 (only lanes 0..3 matter)
  const int h = lane & (NHEAD - 1);
  float e = lrelu(el[s * NHEAD + h] + er[d * NHEAD + h]) * wv;
  const float m = fdec(menc[d * NHEAD + h]);
  const float a = __expf(e - m) / den[d * NHEAD + h];
  const float ah = __shfl(a, lane >> 3, 32);  // head for this lane's 4 comps

  const float4 f = *(const float4*)(feat + s * HD + lane * 4);
  float* __restrict__ op = out + d * HD + lane * 4;
  unsafeAtomicAdd(op + 0, f.x * ah);
  unsafeAtomicAdd(op + 1, f.y * ah);
  unsafeAtomicAdd(op + 2, f.z * ah);
  unsafeAtomicAdd(op + 3, f.w * ah);
}

// =====================================================================
// Host launcher
// =====================================================================
extern "C" void kernel_launch(void* const* d_in, const int* in_sizes, int n_in,
                              void* d_out, int out_size, void* d_ws, size_t ws_size,
                              hipStream_t stream) {
  // setup_inputs order: vt, x, src, dst, w, W_fc, attn_l, attn_r
  const float* x      = (const float*)d_in[1];
  const int*   src    = (const int*)d_in[2];
  const int*   dst    = (const int*)d_in[3];
  const float* w      = (const float*)d_in[4];
  const float* W_fc   = (const float*)d_in[5];
  const float* attn_l = (const float*)d_in[6];
  const float* attn_r = (const float*)d_in[7];
  float* out = (float*)d_out;

  const int N = in_sizes[1] / IN_DIM;
  const int E = in_sizes[2];

  // workspace layout (bytes)
  char* ws = (char*)d_ws;
  size_t off = 0;
  float*    feat = (float*)(ws + off);    off += (size_t)N * HD * sizeof(float);
  float*    el   = (float*)(ws + off);    off += (size_t)N * NHEAD * sizeof(float);
  float*    er   = (float*)(ws + off);    off += (size_t)N * NHEAD * sizeof(float);
  unsigned* menc = (unsigned*)(ws + off); off += (size_t)N * NHEAD * sizeof(unsigned);
  float*    den  = (float*)(ws + off);    off += (size_t)N * NHEAD * sizeof(float);
  (void)ws_size;

  // re-init accumulators every call (graph-capture safe)
  hipMemsetAsync(menc, 0, (size_t)N * NHEAD * sizeof(unsigned), stream); // enc-domain -inf
  hipMemsetAsync(den,  0, (size_t)N * NHEAD * sizeof(float), stream);
  hipMemsetAsync(out,  0, (size_t)out_size * sizeof(float), stream);

  // 1) projection GEMM (WMMA f32): one wave per 16-row tile, 8 waves/block
  const int tiles = (N + 15) / 16;
  const int pblocks = (tiles + 7) / 8;
  gat_proj_wmma<<<pblocks, 256, 0, stream>>>(x, W_fc, feat, N);

  // 2) attention dots
  const int n_nh = N * NHEAD;
  gat_attn_dots<<<(n_nh + 255) / 256, 256, 0, stream>>>(feat, attn_l, attn_r, el, er, n_nh);

  // 3) edge softmax max
  const int n_eh = E * NHEAD;
  gat_edge_max<<<(n_eh + 255) / 256, 256, 0, stream>>>(src, dst, w, el, er, menc, n_eh);

  // 4) edge softmax denominator
  gat_edge_den<<<(n_eh + 255) / 256, 256, 0, stream>>>(src, dst, w, el, er, menc, den, n_eh);

  // 5) weighted message scatter: one wave per edge
  const long long threads = (long long)E * 32;
  gat_edge_scatter<<<(int)((threads + 255) / 256), 256, 0, stream>>>(
      src, dst, w, el, er, menc, den, feat, out, E);
}